// GeneralizedInteractionNet_21852793602482
// MI455X (gfx1250) — compile-verified
//
#include <hip/hip_runtime.h>
#include <hip/hip_bf16.h>

// GeneralizedInteractionNet on gfx1250 (MI455X), fp32 WMMA (V_WMMA_F32_16X16X4_F32)
// + async global<->LDS copies (ASYNCcnt) for staging and writeback.
//
// out[b,o,h] = sum_{i,j,d} B0[b,j,h] * Bi[b,i,d] * alpha[i*24+j,o] * W[o,h,d] * hv[o,d]
// Factored:
//   A[i,o,h] = sum_j alpha[i*24+j,o] * B0[b,j,h]        (WMMA: M=o, N=h, K=24)
//   V[i,N]   = sum_d Bi[b,i,d] * Wh[N,d], N=o*32+h      (WMMA: M=i, N=512, K=32)
//   out[b,N] = sum_i A[i,N] * V[i,N]                    (fused epilogue, ds_add_f32)
//
// One wave32 per batch element; 3 dependent launches (s_in = 24,16,16).

typedef __attribute__((ext_vector_type(2))) float v2f;
typedef __attribute__((ext_vector_type(8))) float v8f;

#define NF    24    // num_fields
#define D     32    // emb
#define NSUB  16    // output subspaces
#define ON    512   // 16*32 output elems per batch
#define WHS   36    // padded d-stride of Wh rows (bank-conflict avoidance, 16B aligned)
#define AOS   33    // padded h-stride of A rows  (lane-half store conflict avoidance)

__device__ __forceinline__ void async_ld_b128(unsigned lds_addr, const void* gaddr) {
    asm volatile("global_load_async_to_lds_b128 %0, %1, off"
                 :: "v"(lds_addr), "v"((unsigned long long)(uintptr_t)gaddr)
                 : "memory");
}
__device__ __forceinline__ void async_st_b128(void* gaddr, unsigned lds_addr) {
    asm volatile("global_store_async_from_lds_b128 %0, %1, off"
                 :: "v"((unsigned long long)(uintptr_t)gaddr), "v"(lds_addr)
                 : "memory");
}
__device__ __forceinline__ void wait_async0() {
    asm volatile("s_wait_asynccnt 0x0" ::: "memory");
}
__device__ __forceinline__ void wait_ds0() {
    asm volatile("s_wait_dscnt 0x0" ::: "memory");
}

template <int S_IN>
__global__ __launch_bounds__(32)
void gin_layer(const float* __restrict__ B0,     // (512, 24, 32)
               const float* __restrict__ Bi,     // (512, S_IN, 32)
               const float* __restrict__ alpha,  // (S_IN*24, 16)
               const float* __restrict__ W,      // (16, 32, 32)
               const float* __restrict__ hv,     // (16, 32, 1)
               float* __restrict__ out)          // (512, 16, 32)
{
    // LDS layout (floats); every region 16B-aligned.
    constexpr int OFF_WH  = 0;                         // Wh[o][h][dpad]: 16*32*36
    constexpr int OFF_A   = OFF_WH + NSUB * D * WHS;   // A[i][o][hpad] : S_IN*16*33
    constexpr int OFF_B0  = OFF_A  + S_IN * NSUB * AOS + ((S_IN * NSUB * AOS) % 4 ? 4 - (S_IN * NSUB * AOS) % 4 : 0);
    constexpr int OFF_BI  = OFF_B0 + NF * D;           // Bi_b : S_IN*32
    constexpr int OFF_OUT = OFF_BI + S_IN * D;         // out-acc : 512
    constexpr int LDS_FLOATS = OFF_OUT + ON;

    extern __shared__ float lds[];

    const int l  = threadIdx.x;   // 0..31
    const int lm = l & 15;        // lane % 16
    const int lh = l >> 4;        // lane half (0/1)
    const int b  = blockIdx.x;

    // ---- async-stage B0_b and Bi_b (global -> LDS, bypassing VGPRs) ----
#pragma unroll
    for (int k = 0; k < (NF * D) / (4 * 32); ++k) {            // 6 x b128 per lane
        const int q = l + 32 * k;                              // 16B quad index
        async_ld_b128((unsigned)(uintptr_t)&lds[OFF_B0 + 4 * q],
                      B0 + b * (NF * D) + 4 * q);
    }
#pragma unroll
    for (int k = 0; k < (S_IN * D) / (4 * 32); ++k) {
        const int q = l + 32 * k;
        async_ld_b128((unsigned)(uintptr_t)&lds[OFF_BI + 4 * q],
                      Bi + b * (S_IN * D) + 4 * q);
    }

    // ---- stage Wh = W * hv into padded LDS (float4), overlap with async ----
    {
        const float4* W4  = reinterpret_cast<const float4*>(W);
        const float4* hv4 = reinterpret_cast<const float4*>(hv);
        for (int t4 = l; t4 < (NSUB * D * D) / 4; t4 += 32) {
            const int t  = 4 * t4;              // = o*1024 + h*32 + d, d % 4 == 0
            const int o  = t >> 10;
            const int hh = (t >> 5) & 31;
            const int d  = t & 31;
            float4 w = W4[t4];
            float4 s = hv4[((o << 5) + d) >> 2];
            w.x *= s.x; w.y *= s.y; w.z *= s.z; w.w *= s.w;
            *reinterpret_cast<float4*>(&lds[OFF_WH + o * (D * WHS) + hh * WHS + d]) = w;
        }
        for (int t = l; t < ON; t += 32) lds[OFF_OUT + t] = 0.0f;
    }
    wait_async0();
    __syncthreads();

    // ---- A-step: A[i] = alpha_i^T (16x24) x B0_b (24x32) ----
    for (int i = 0; i < S_IN; ++i) {
        v2f av[6];
#pragma unroll
        for (int k = 0; k < 6; ++k) {           // A operand: M=o=lm, K=kk+v
            const int kk = 4 * k + lh * 2;
            av[k].x = alpha[(i * NF + kk    ) * NSUB + lm];
            av[k].y = alpha[(i * NF + kk + 1) * NSUB + lm];
        }
        v8f c0 = {}, c1 = {};
#pragma unroll
        for (int k = 0; k < 6; ++k) {           // B operand: K=j, N=h (two h-tiles)
            const int kk = 4 * k + lh * 2;
            v2f b0t, b1t;
            b0t.x = lds[OFF_B0 + (kk    ) * D + lm];
            b0t.y = lds[OFF_B0 + (kk + 1) * D + lm];
            b1t.x = lds[OFF_B0 + (kk    ) * D + 16 + lm];
            b1t.y = lds[OFF_B0 + (kk + 1) * D + 16 + lm];
            c0 = __builtin_amdgcn_wmma_f32_16x16x4_f32(false, av[k], false, b0t, (short)0, c0, false, false);
            c1 = __builtin_amdgcn_wmma_f32_16x16x4_f32(false, av[k], false, b1t, (short)0, c1, false, false);
        }
#pragma unroll
        for (int r = 0; r < 8; ++r) {           // C: lane->N=h, reg->M=o (+8 per half)
            const int o = r + lh * 8;
            lds[OFF_A + i * (NSUB * AOS) + o * AOS + lm]      = c0[r];
            lds[OFF_A + i * (NSUB * AOS) + o * AOS + 16 + lm] = c1[r];
        }
    }
    __syncthreads();

    // ---- V-step + fused i-reduction: V = Bi_b (S_INx32) x Wh^T (32x512) ----
    constexpr int MT = (S_IN + 15) / 16;
    for (int mt = 0; mt < MT; ++mt) {
        const int   i0    = mt * 16 + lm;
        const float amask = (i0 < S_IN) ? 1.0f : 0.0f;
        const int   ia    = (i0 < S_IN) ? i0 : (S_IN - 1);
        v2f av[8];
#pragma unroll
        for (int k = 0; k < 8; ++k) {           // A operand: M=i, K=d (hoisted over nt)
            const int kk = 4 * k + lh * 2;
            av[k].x = amask * lds[OFF_BI + ia * D + kk];
            av[k].y = amask * lds[OFF_BI + ia * D + kk + 1];
        }
        for (int nt = 0; nt < 32; ++nt) {
            const int N    = nt * 16 + lm;      // N = o*32 + h
            const int o    = N >> 5;
            const int hh   = N & 31;
            const int base = OFF_WH + o * (D * WHS) + hh * WHS;
            v8f c = {};
#pragma unroll
            for (int k = 0; k < 8; ++k) {       // B operand: K=d, N
                const int kk = 4 * k + lh * 2;
                v2f bb;
                bb.x = lds[base + kk];
                bb.y = lds[base + kk + 1];
                c = __builtin_amdgcn_wmma_f32_16x16x4_f32(false, av[k], false, bb, (short)0, c, false, false);
            }
            float partial = 0.0f;
#pragma unroll
            for (int r = 0; r < 8; ++r) {       // D row = i index
                const int   i  = mt * 16 + lh * 8 + r;
                const float m2 = (i < S_IN) ? 1.0f : 0.0f;
                const int   ic = (i < S_IN) ? i : (S_IN - 1);
                partial += m2 * c[r] * lds[OFF_A + ic * (NSUB * AOS) + o * AOS + hh];
            }
            atomicAdd(&lds[OFF_OUT + N], partial);             // ds_add_f32
        }
    }
    wait_ds0();
    __syncthreads();

    // ---- async writeback: LDS out-acc -> global ----
#pragma unroll
    for (int k = 0; k < ON / (4 * 32); ++k) {                  // 4 x b128 per lane
        const int q = l + 32 * k;
        async_st_b128(out + b * ON + 4 * q,
                      (unsigned)(uintptr_t)&lds[OFF_OUT + 4 * q]);
    }
    wait_async0();
    (void)LDS_FLOATS;
}

static constexpr size_t lds_bytes(int s_in) {
    size_t a = (size_t)NSUB * D * WHS + (size_t)s_in * NSUB * AOS;
    a += (a % 4) ? (4 - a % 4) : 0;
    a += (size_t)NF * D + (size_t)s_in * D + ON;
    return a * sizeof(float);
}

extern "C" void kernel_launch(void* const* d_in, const int* in_sizes, int n_in,
                              void* d_out, int out_size, void* d_ws, size_t ws_size,
                              hipStream_t stream) {
    (void)in_sizes; (void)n_in; (void)out_size; (void)ws_size;
    const float* B0 = (const float*)d_in[0];
    const float* W0 = (const float*)d_in[1];
    const float* a0 = (const float*)d_in[2];
    const float* h0 = (const float*)d_in[3];
    const float* W1 = (const float*)d_in[4];
    const float* a1 = (const float*)d_in[5];
    const float* h1 = (const float*)d_in[6];
    const float* W2 = (const float*)d_in[7];
    const float* a2 = (const float*)d_in[8];
    const float* h2 = (const float*)d_in[9];

    float* Bi1 = (float*)d_ws;                 // 512*512 floats = 1 MB
    float* Bi2 = Bi1 + 512 * ON;               // second 1 MB of scratch
    float* O   = (float*)d_out;

    gin_layer<24><<<512, 32, lds_bytes(24), stream>>>(B0, B0,  a0, W0, h0, Bi1);
    gin_layer<16><<<512, 32, lds_bytes(16), stream>>>(B0, Bi1, a1, W1, h1, Bi2);
    gin_layer<16><<<512, 32, lds_bytes(16), stream>>>(B0, Bi2, a2, W2, h2, O);
}